// OpTA_1829656068561
// MI455X (gfx1250) — compile-verified
//
#include <hip/hip_runtime.h>
#include <hip/hip_bf16.h>
#include <math.h>

// ---------------------------------------------------------------------------
// Sinkhorn optimal transport on MI455X (gfx1250, wave32).
//  zs:[1024,256] f32, zq:[8192,256] f32
//  C = x2 + y2 - 2*zs@zq^T          (V_WMMA_F32_16X16X4_F32 GEMM)
//  100x log-domain Sinkhorn steps with device-side convergence flag
//  plan = exp(M)/rowsum ; out0 = plan @ zq (split-K WMMA GEMM) ; out1 = zq
// ---------------------------------------------------------------------------

#define NS 1024
#define NQ 8192
#define DD 256
#define EPS_R 0.05f
#define INV_EPS 20.0f
#define THRESH 1e-3f
#define MAX_ITER 100
#define KCH 8              // split-K chunks for gemm2 / i-chunks for v-update

typedef float v2f __attribute__((ext_vector_type(2)));
typedef float v8f __attribute__((ext_vector_type(8)));

__device__ __forceinline__ v8f wmma_f32_16x16x4(v2f a, v2f b, v8f c) {
  // 8-arg form: (neg_a, A, neg_b, B, c_mod, C, reuse_a, reuse_b)
  return __builtin_amdgcn_wmma_f32_16x16x4_f32(false, a, false, b, (short)0, c,
                                               false, false);
}

// online logsumexp combine: (m,s) <- combine((m,s),(m2,s2)), all finite
__device__ __forceinline__ void lse_comb(float& m, float& s, float m2, float s2) {
  float mn = fmaxf(m, m2);
  s = s * __expf(m - mn) + s2 * __expf(m2 - mn);
  m = mn;
}

__device__ __forceinline__ void wave_lse_reduce(float& m, float& s) {
#pragma unroll
  for (int off = 16; off > 0; off >>= 1) {
    float m2 = __shfl_xor(m, off, 32);
    float s2 = __shfl_xor(s, off, 32);
    lse_comb(m, s, m2, s2);
  }
}

// ---------------------------------------------------------------------------
// squared row norms: one wave per row (zs rows first, then zq rows)
__global__ void sqnorm_kernel(const float* __restrict__ zs,
                              const float* __restrict__ zq,
                              float* __restrict__ x2, float* __restrict__ y2) {
  int row = blockIdx.x * 8 + (threadIdx.x >> 5);
  int lane = threadIdx.x & 31;
  const float* p;
  float* outp;
  if (row < NS) { p = zs + row * DD; outp = x2 + row; }
  else          { p = zq + (row - NS) * DD; outp = y2 + (row - NS); }
  float s = 0.f;
  for (int k = lane; k < DD; k += 32) { float t = p[k]; s += t * t; }
#pragma unroll
  for (int off = 16; off > 0; off >>= 1) s += __shfl_xor(s, off, 32);
  if (lane == 0) *outp = s;
}

__global__ void init_kernel(float* __restrict__ u, float* __restrict__ v,
                            float* __restrict__ du, int* __restrict__ done) {
  int t = blockIdx.x * 256 + threadIdx.x;
  if (t < NS) { u[t] = 0.f; du[t] = 0.f; }
  if (t < NQ) v[t] = 0.f;
  if (t == 0) *done = 0;
}

// ---------------------------------------------------------------------------
// GEMM1: C[m,n] = x2[m] + y2[n] - 2 * sum_k zs[m,k]*zq[n,k]
// wave tile 16Mx64N, block = 8 waves = 32Mx256N.  grid (NQ/256, NS/32)
__global__ void cost_kernel(const float* __restrict__ zs,
                            const float* __restrict__ zq,
                            const float* __restrict__ x2,
                            const float* __restrict__ y2,
                            float* __restrict__ Cm) {
  const int wave = threadIdx.x >> 5, lane = threadIdx.x & 31;
  const int wm = wave >> 2, wn = wave & 3;
  const int m0 = blockIdx.y * 32 + wm * 16;
  const int n0 = blockIdx.x * 256 + wn * 64;
  const int ln = lane & 15, kh = lane >> 4;   // A/B lane decode

  v8f acc[4] = {};
  const float* ap = zs + (m0 + ln) * DD + 2 * kh;
  for (int k = 0; k < DD; k += 4) {
    v2f a = *(const v2f*)(ap + k);
#pragma unroll
    for (int t = 0; t < 4; ++t) {
      // B = zq^T : B[k][n] = zq[n][k]  -> contiguous pair per lane
      v2f b = *(const v2f*)(zq + (n0 + t * 16 + ln) * DD + k + 2 * kh);
      acc[t] = wmma_f32_16x16x4(a, b, acc[t]);
    }
  }
  // D layout: VGPR r -> M = m0 + r + 8*(lane>=16); N = n0 + t*16 + (lane&15)
#pragma unroll
  for (int t = 0; t < 4; ++t)
#pragma unroll
    for (int r = 0; r < 8; ++r) {
      int m = m0 + r + 8 * kh;
      int n = n0 + t * 16 + ln;
      Cm[m * NQ + n] = x2[m] + y2[n] - 2.0f * acc[t][r];
    }
}

// ---------------------------------------------------------------------------
// u-update: one block per row i; coalesced row scan + block logsumexp
__global__ void u_step_kernel(const float* __restrict__ Cm,
                              float* __restrict__ u,
                              const float* __restrict__ v,
                              float* __restrict__ du,
                              const int* __restrict__ done) {
  if (*done) return;
  const int i = blockIdx.x;
  const float ui = u[i];
  const float* row = Cm + i * NQ;
  int t = threadIdx.x;
  float z = (ui + v[t] - row[t]) * INV_EPS;
  float m = z, s = 1.0f;
  for (int j = t + 256; j < NQ; j += 256) {
    z = (ui + v[j] - row[j]) * INV_EPS;
    float mn = fmaxf(m, z);
    s = s * __expf(m - mn) + __expf(z - mn);
    m = mn;
  }
  wave_lse_reduce(m, s);
  __shared__ float sm[8], ss[8];
  int wave = threadIdx.x >> 5, lane = threadIdx.x & 31;
  if (lane == 0) { sm[wave] = m; ss[wave] = s; }
  __syncthreads();
  if (threadIdx.x == 0) {
    float M = sm[0], S = ss[0];
#pragma unroll
    for (int w = 1; w < 8; ++w) lse_comb(M, S, sm[w], ss[w]);
    float lse = M + __logf(S);
    float log_mu = __logf(1.0f / NS + 1e-8f);
    float un = EPS_R * (log_mu - lse) + ui;
    du[i] = fabsf(un - ui);
    u[i] = un;
  }
}

// v-update stage A: thread-per-column over an i-chunk (coalesced C reads)
// grid (NQ/256, KCH)
__global__ void v_partial_kernel(const float* __restrict__ Cm,
                                 const float* __restrict__ u,
                                 const float* __restrict__ v,
                                 float* __restrict__ pm, float* __restrict__ ps,
                                 const int* __restrict__ done) {
  if (*done) return;
  const int j = blockIdx.x * 256 + threadIdx.x;
  const int c = blockIdx.y;
  const int i0 = c * (NS / KCH);
  const float vj = v[j];
  float z = (u[i0] + vj - Cm[i0 * NQ + j]) * INV_EPS;
  float m = z, s = 1.0f;
  for (int i = i0 + 1; i < i0 + NS / KCH; ++i) {
    z = (u[i] + vj - Cm[i * NQ + j]) * INV_EPS;
    float mn = fmaxf(m, z);
    s = s * __expf(m - mn) + __expf(z - mn);
    m = mn;
  }
  pm[c * NQ + j] = m;
  ps[c * NQ + j] = s;
}

// v-update stage B: deterministic combine of the KCH partials
__global__ void v_update_kernel(const float* __restrict__ pm,
                                const float* __restrict__ ps,
                                float* __restrict__ v,
                                const int* __restrict__ done) {
  if (*done) return;
  const int j = blockIdx.x * 256 + threadIdx.x;
  float m = pm[j], s = ps[j];
#pragma unroll
  for (int c = 1; c < KCH; ++c) lse_comb(m, s, pm[c * NQ + j], ps[c * NQ + j]);
  float lse = m + __logf(s);
  float log_nu = __logf(1.0f / NQ + 1e-8f);
  v[j] = EPS_R * (log_nu - lse) + v[j];
}

// err = sum|du|; set done if below threshold (deterministic tree sum)
__global__ void end_iter_kernel(const float* __restrict__ du,
                                int* __restrict__ done) {
  if (*done) return;
  __shared__ float red[256];
  float s = 0.f;
  for (int i = threadIdx.x; i < NS; i += 256) s += du[i];
  red[threadIdx.x] = s;
  __syncthreads();
#pragma unroll
  for (int off = 128; off > 0; off >>= 1) {
    if (threadIdx.x < off) red[threadIdx.x] += red[threadIdx.x + off];
    __syncthreads();
  }
  if (threadIdx.x == 0 && red[0] < THRESH) *done = 1;
}

// ---------------------------------------------------------------------------
// rs[i] = logsumexp_j M(i,j)   (log of the row sum of the transport plan)
__global__ void rowlse_kernel(const float* __restrict__ Cm,
                              const float* __restrict__ u,
                              const float* __restrict__ v,
                              float* __restrict__ rs) {
  const int i = blockIdx.x;
  const float ui = u[i];
  const float* row = Cm + i * NQ;
  int t = threadIdx.x;
  float z = (ui + v[t] - row[t]) * INV_EPS;
  float m = z, s = 1.0f;
  for (int j = t + 256; j < NQ; j += 256) {
    z = (ui + v[j] - row[j]) * INV_EPS;
    float mn = fmaxf(m, z);
    s = s * __expf(m - mn) + __expf(z - mn);
    m = mn;
  }
  wave_lse_reduce(m, s);
  __shared__ float sm[8], ss[8];
  int wave = threadIdx.x >> 5, lane = threadIdx.x & 31;
  if (lane == 0) { sm[wave] = m; ss[wave] = s; }
  __syncthreads();
  if (threadIdx.x == 0) {
    float M = sm[0], S = ss[0];
#pragma unroll
    for (int w = 1; w < 8; ++w) lse_comb(M, S, sm[w], ss[w]);
    rs[i] = M + __logf(S);
  }
}

// plan_norm[i,j] = exp(M(i,j) - rs[i])   (overwrites C in place)
__global__ void plan_kernel(float* __restrict__ Cm, const float* __restrict__ u,
                            const float* __restrict__ v,
                            const float* __restrict__ rs) {
  int idx = blockIdx.x * 256 + threadIdx.x;   // < NS*NQ = 8388608
  int i = idx >> 13;                          // / NQ
  int j = idx & (NQ - 1);
  float M = (u[i] + v[j] - Cm[idx]) * INV_EPS;
  Cm[idx] = __expf(M - rs[i]);
}

// ---------------------------------------------------------------------------
// GEMM2 (split-K): partial[kc] += plan[m, k0:k0+1024] @ zq[k0:k0+1024, :]
// grid (NS/32, KCH); block 8 waves (2Mx4N), wave tile 16Mx64N, N fully covered
__global__ void gemm2_kernel(const float* __restrict__ plan,
                             const float* __restrict__ zq,
                             float* __restrict__ partial) {
  const int wave = threadIdx.x >> 5, lane = threadIdx.x & 31;
  const int wm = wave >> 2, wn = wave & 3;
  const int m0 = blockIdx.x * 32 + wm * 16;
  const int n0 = wn * 64;
  const int kc = blockIdx.y;
  const int k0 = kc * (NQ / KCH);
  const int ln = lane & 15, kh = lane >> 4;

  v8f acc[4] = {};
  const float* ap = plan + (m0 + ln) * NQ + 2 * kh;
  for (int k = k0; k < k0 + NQ / KCH; k += 4) {
    v2f a = *(const v2f*)(ap + k);
    const float* b0 = zq + (k + 2 * kh) * DD + n0 + ln;
#pragma unroll
    for (int t = 0; t < 4; ++t) {
      v2f b;
      b.x = b0[t * 16];         // B[k+2kh  ][n]
      b.y = b0[DD + t * 16];    // B[k+2kh+1][n]
      acc[t] = wmma_f32_16x16x4(a, b, acc[t]);
    }
  }
#pragma unroll
  for (int t = 0; t < 4; ++t)
#pragma unroll
    for (int r = 0; r < 8; ++r) {
      int m = m0 + r + 8 * kh;
      int n = n0 + t * 16 + ln;
      partial[(kc * NS + m) * DD + n] = acc[t][r];
    }
}

__global__ void reduce_out_kernel(const float* __restrict__ partial,
                                  float* __restrict__ out) {
  int idx = blockIdx.x * 256 + threadIdx.x;   // < NS*DD = 262144
  float s = 0.f;
#pragma unroll
  for (int c = 0; c < KCH; ++c) s += partial[c * NS * DD + idx];
  out[idx] = s;
}

__global__ void copy_zq_kernel(const float4* __restrict__ src,
                               float4* __restrict__ dst) {
  int idx = blockIdx.x * 256 + threadIdx.x;   // < NQ*DD/4 = 524288
  dst[idx] = src[idx];
}

// ---------------------------------------------------------------------------
extern "C" void kernel_launch(void* const* d_in, const int* in_sizes, int n_in,
                              void* d_out, int out_size, void* d_ws,
                              size_t ws_size, hipStream_t stream) {
  const float* zs = (const float*)d_in[0];   // [1024,256]
  const float* zq = (const float*)d_in[1];   // [8192,256]
  float* out = (float*)d_out;                // [1024*256 + 8192*256]

  // workspace layout (floats)
  float* Cm   = (float*)d_ws;                 // NS*NQ        (32 MB)
  float* pm   = Cm + (size_t)NS * NQ;         // KCH*NQ
  float* ps   = pm + KCH * NQ;                // KCH*NQ
  float* part = ps + KCH * NQ;                // KCH*NS*DD    (8 MB)
  float* x2   = part + KCH * NS * DD;         // NS
  float* y2   = x2 + NS;                      // NQ
  float* u    = y2 + NQ;                      // NS
  float* v    = u + NS;                       // NQ
  float* du   = v + NQ;                       // NS
  float* rs   = du + NS;                      // NS
  int*   done = (int*)(rs + NS);              // 1

  sqnorm_kernel<<<(NS + NQ) / 8, 256, 0, stream>>>(zs, zq, x2, y2);
  init_kernel<<<NQ / 256, 256, 0, stream>>>(u, v, du, done);
  cost_kernel<<<dim3(NQ / 256, NS / 32), 256, 0, stream>>>(zs, zq, x2, y2, Cm);

  for (int it = 0; it < MAX_ITER; ++it) {
    u_step_kernel<<<NS, 256, 0, stream>>>(Cm, u, v, du, done);
    v_partial_kernel<<<dim3(NQ / 256, KCH), 256, 0, stream>>>(Cm, u, v, pm, ps,
                                                              done);
    v_update_kernel<<<NQ / 256, 256, 0, stream>>>(pm, ps, v, done);
    end_iter_kernel<<<1, 256, 0, stream>>>(du, done);
  }

  rowlse_kernel<<<NS, 256, 0, stream>>>(Cm, u, v, rs);
  plan_kernel<<<(NS * NQ) / 256, 256, 0, stream>>>(Cm, u, v, rs);
  gemm2_kernel<<<dim3(NS / 32, KCH), 256, 0, stream>>>(Cm, zq, part);
  reduce_out_kernel<<<(NS * DD) / 256, 256, 0, stream>>>(part, out);
  copy_zq_kernel<<<(NQ * DD / 4) / 256, 256, 0, stream>>>(
      (const float4*)zq, (float4*)(out + NS * DD));
}